// MultiheadAttention_52115133170084
// MI455X (gfx1250) — compile-verified
//
#include <hip/hip_runtime.h>
#include <math.h>

typedef __attribute__((ext_vector_type(2))) float v2f;
typedef __attribute__((ext_vector_type(8))) float v8f;

#define T_DIM 2048
#define B_DIM 2
#define E_DIM 512
#define H_DIM 8
#define HD_DIM 64
#define BH_DIM 16
#define ROWS (T_DIM * B_DIM)                 // 4096
#define LOGD (-0.05129329438755058f)         // ln(0.95)
#define INV_T (1.0f / 2048.0f)
#define KVPAD 68                             // 64 + 4 pad floats per LDS row

static __device__ __forceinline__ v8f wmma4(v2f a, v2f b, v8f c) {
  // D = A(16x4,f32) * B(4x16,f32) + C(16x16,f32)
  return __builtin_amdgcn_wmma_f32_16x16x4_f32(false, a, false, b, (short)0, c,
                                               false, false);
}

static __device__ __forceinline__ float rmax16(float v) {
#pragma unroll
  for (int m = 1; m <= 8; m <<= 1) v = fmaxf(v, __shfl_xor(v, m, 32));
  return v;
}
static __device__ __forceinline__ float rsum16(float v) {
#pragma unroll
  for (int m = 1; m <= 8; m <<= 1) v += __shfl_xor(v, m, 32);
  return v;
}

// Cooperative async stage of one 16x64 f32 tile (row stride B_DIM*E_DIM floats
// in global, KVPAD floats in LDS). 8 x global_load_async_to_lds_b128 per wave:
// each op moves 2 rows (16 lanes x 16B per row). Tracked by ASYNCcnt.
static __device__ __forceinline__ void stage16x64(float* dst, const float* src,
                                                  int half, int l15) {
#pragma unroll
  for (int i = 0; i < 8; ++i) {
    const int r = i * 2 + half;
    uint32_t la = (uint32_t)(uintptr_t)(dst + r * KVPAD + l15 * 4);
    const float* ga = src + (size_t)r * ((size_t)B_DIM * E_DIM) + l15 * 4;
    asm volatile("global_load_async_to_lds_b128 %0, %1, off" ::"v"(la), "v"(ga)
                 : "memory");
  }
}

// ---------------------------------------------------------------------------
// Kernel A: q/k/v projections. Y = X @ W^T + b  (q additionally * HD^-0.5)
// One wave per 16x16 output tile. X: (4096, 512) row-major.
// ---------------------------------------------------------------------------
__global__ __launch_bounds__(128) void proj_qkv_kernel(
    const float* __restrict__ X, const float* __restrict__ Wq,
    const float* __restrict__ bq, const float* __restrict__ Wk,
    const float* __restrict__ bk, const float* __restrict__ Wv,
    const float* __restrict__ bv, float* __restrict__ q_out,
    float* __restrict__ k_out, float* __restrict__ v_out) {
  const int wave = blockIdx.x * 4 + (threadIdx.x >> 5);
  const int lane = threadIdx.x & 31;
  const int tilesPerProj = (ROWS / 16) * (E_DIM / 16);  // 8192
  const int proj = wave / tilesPerProj;
  const int rem = wave % tilesPerProj;
  const int m0 = (rem / (E_DIM / 16)) * 16;
  const int n0 = (rem % (E_DIM / 16)) * 16;

  const float* W = (proj == 0) ? Wq : (proj == 1) ? Wk : Wv;
  const float* bias = (proj == 0) ? bq : (proj == 1) ? bk : bv;
  float* Y = (proj == 0) ? q_out : (proj == 1) ? k_out : v_out;
  const float scale = (proj == 0) ? 0.125f : 1.0f;  // HD^-0.5 = 1/8

  const int half = lane >> 4;
  const int l15 = lane & 15;
  const float* xrow = X + (size_t)(m0 + l15) * E_DIM + 2 * half;
  const float* wrow = W + (size_t)(n0 + l15) * E_DIM + 2 * half;

  v8f acc = {0.f, 0.f, 0.f, 0.f, 0.f, 0.f, 0.f, 0.f};
#pragma unroll 8
  for (int k = 0; k < E_DIM; k += 4) {
    v2f a = *(const v2f*)(xrow + k);
    v2f b = *(const v2f*)(wrow + k);
    acc = wmma4(a, b, acc);
  }
#pragma unroll
  for (int j = 0; j < 8; ++j) {
    const int row = m0 + j + 8 * half;
    const int col = n0 + l15;
    Y[(size_t)row * E_DIM + col] = (acc[j] + bias[col]) * scale;
  }
}

// ---------------------------------------------------------------------------
// Kernel B: streaming attention with decay mask. One wave per (bh, 16-row tile)
// out_t = (A_t + (Z_t/T) v_t) / (a_t + Z_t/T)   (softmax Z cancels)
// K/V tiles double-buffered through LDS via async global->LDS DMA.
// ---------------------------------------------------------------------------
__global__ __launch_bounds__(64) void attn_kernel(
    const float* __restrict__ Q, const float* __restrict__ K,
    const float* __restrict__ V, const float* __restrict__ bias,
    float* __restrict__ Aout) {
  __shared__ float kbuf[2][2][16 * KVPAD];  // [wave][buf][tile]
  __shared__ float vbuf[2][2][16 * KVPAD];
  __shared__ float plds[2][16 * 17];        // per-wave P-tile staging (padded)
  const int wslot = threadIdx.x >> 5;
  const int lane = threadIdx.x & 31;
  const int wave = blockIdx.x * 2 + wslot;
  const int bh = wave / (T_DIM / 16);
  const int tt = wave % (T_DIM / 16);
  const int b = bh / H_DIM, h = bh % H_DIM;
  const int t0 = tt * 16;
  const int half = lane >> 4, l15 = lane & 15;

  const size_t rstride = (size_t)B_DIM * E_DIM;  // floats between rows of t
  const float* Qb = Q + (size_t)b * E_DIM + h * HD_DIM;
  const float* Kb = K + (size_t)b * E_DIM + h * HD_DIM;
  const float* Vb = V + (size_t)b * E_DIM + h * HD_DIM;
  const float* biasB = bias + (size_t)bh * T_DIM * T_DIM;
  float* P = plds[wslot];

  // Preload Q A-fragments for all 16 k-steps of HD=64
  v2f qa[16];
  {
    const float* qrow = Qb + (size_t)(t0 + l15) * rstride + 2 * half;
#pragma unroll
    for (int kk = 0; kk < 16; ++kk) qa[kk] = *(const v2f*)(qrow + kk * 4);
  }

  float m[8], zp[8], ap[8];
  v8f acc[4];
#pragma unroll
  for (int j = 0; j < 8; ++j) {
    m[j] = -3.0e38f;
    zp[j] = 0.f;
    ap[j] = 0.f;
  }
#pragma unroll
  for (int u = 0; u < 4; ++u)
    acc[u] = (v8f){0.f, 0.f, 0.f, 0.f, 0.f, 0.f, 0.f, 0.f};

  // Prime the pipeline: stage tile 0 (16 async ops outstanding)
  stage16x64(kbuf[wslot][0], Kb, half, l15);
  stage16x64(vbuf[wslot][0], Vb, half, l15);

  for (int st = 0; st < T_DIM / 16; ++st) {
    const int s0 = st * 16;
    const int cur = st & 1;
    float* kcur = kbuf[wslot][cur];
    float* vcur = vbuf[wslot][cur];
    // Stage next tile into the other buffer, then wait for the current tile.
    // Async loads retire in order: <=16 outstanding => tile st has landed.
    if (st + 1 < T_DIM / 16) {
      const float* knext = Kb + (size_t)(st + 1) * 16 * rstride;
      const float* vnext = Vb + (size_t)(st + 1) * 16 * rstride;
      stage16x64(kbuf[wslot][cur ^ 1], knext, half, l15);
      stage16x64(vbuf[wslot][cur ^ 1], vnext, half, l15);
      asm volatile("s_wait_asynccnt 0x10" ::: "memory");
    } else {
      asm volatile("s_wait_asynccnt 0x0" ::: "memory");
    }

    // ---- scores S = q_tile @ k_tile^T (K fragments from LDS) -------------
    v8f S = {0.f, 0.f, 0.f, 0.f, 0.f, 0.f, 0.f, 0.f};
#pragma unroll
    for (int kk = 0; kk < 16; ++kk) {
      v2f kb = *(const v2f*)(kcur + l15 * KVPAD + kk * 4 + 2 * half);
      S = wmma4(qa[kk], kb, S);
    }
    // ---- + bias (the HBM-bound stream, coalesced in C layout) ------------
#pragma unroll
    for (int j = 0; j < 8; ++j)
      S[j] += biasB[(size_t)(t0 + j + 8 * half) * T_DIM + s0 + l15];
    // prefetch next bias tiles: lanes 0-15 cover the 16 rows of tile st+1
    // (one 64B line each), lanes 16-31 reach one tile further ahead
    if (st + 1 < T_DIM / 16)
      __builtin_prefetch(
          &biasB[(size_t)(t0 + l15) * T_DIM + s0 + 16 + half * 16], 0, 1);

    // ---- running max & rescale -------------------------------------------
#pragma unroll
    for (int j = 0; j < 8; ++j) {
      const float mn = fmaxf(m[j], rmax16(S[j]));
      const float f = __expf(m[j] - mn);
      m[j] = mn;
      zp[j] *= f;
      ap[j] *= f;
      acc[0][j] *= f;
      acc[1][j] *= f;
      acc[2][j] *= f;
      acc[3][j] *= f;
    }
    // ---- exp (full row -> Z) and causal decay weights (-> LDS P tile) ----
#pragma unroll
    for (int j = 0; j < 8; ++j) {
      const int row = t0 + j + 8 * half;
      const int diff = row - (s0 + l15);
      const float x = S[j] - m[j];
      zp[j] += __expf(x);
      const float w = (diff >= 0) ? __expf(x + (float)diff * LOGD) : 0.f;
      ap[j] += w;
      P[(j + 8 * half) * 17 + l15] = w;
    }
    // ---- acc += P @ V (P re-read in A-frag layout; V B-frags from LDS) ---
#pragma unroll
    for (int kk = 0; kk < 16; kk += 4) {
      const int kb2 = kk + 2 * half;
      v2f pa;
      pa.x = P[l15 * 17 + kb2];
      pa.y = P[l15 * 17 + kb2 + 1];
      const float* v0 = vcur + kb2 * KVPAD;
      const float* v1 = v0 + KVPAD;
#pragma unroll
      for (int u = 0; u < 4; ++u) {
        v2f vb;
        vb.x = v0[u * 16 + l15];
        vb.y = v1[u * 16 + l15];
        acc[u] = wmma4(pa, vb, acc[u]);
      }
    }
  }

  // ---- finalize: out = (A + (Z/T) v_t) / (a + Z/T) ------------------------
  float zt[8], at[8];
#pragma unroll
  for (int j = 0; j < 8; ++j) {
    zt[j] = rsum16(zp[j]);
    at[j] = rsum16(ap[j]);
  }
  float* Ao = Aout + (size_t)b * E_DIM + h * HD_DIM;
#pragma unroll
  for (int u = 0; u < 4; ++u) {
#pragma unroll
    for (int j = 0; j < 8; ++j) {
      const int row = t0 + j + 8 * half;
      const int col = u * 16 + l15;
      const float vd = Vb[(size_t)row * rstride + col];
      const float zi = zt[j] * INV_T;
      Ao[(size_t)row * rstride + col] = (acc[u][j] + zi * vd) / (at[j] + zi);
    }
  }
}

// ---------------------------------------------------------------------------
// Kernel C: output projection. out = attn @ Wo^T + bo
// ---------------------------------------------------------------------------
__global__ __launch_bounds__(128) void proj_out_kernel(
    const float* __restrict__ Xa, const float* __restrict__ Wo,
    const float* __restrict__ bo, float* __restrict__ Y) {
  const int wave = blockIdx.x * 4 + (threadIdx.x >> 5);
  const int lane = threadIdx.x & 31;
  const int m0 = (wave / (E_DIM / 16)) * 16;
  const int n0 = (wave % (E_DIM / 16)) * 16;
  const int half = lane >> 4, l15 = lane & 15;
  const float* xrow = Xa + (size_t)(m0 + l15) * E_DIM + 2 * half;
  const float* wrow = Wo + (size_t)(n0 + l15) * E_DIM + 2 * half;
  v8f acc = {0.f, 0.f, 0.f, 0.f, 0.f, 0.f, 0.f, 0.f};
#pragma unroll 8
  for (int k = 0; k < E_DIM; k += 4)
    acc = wmma4(*(const v2f*)(xrow + k), *(const v2f*)(wrow + k), acc);
#pragma unroll
  for (int j = 0; j < 8; ++j)
    Y[(size_t)(m0 + j + 8 * half) * E_DIM + n0 + l15] = acc[j] + bo[n0 + l15];
}

extern "C" void kernel_launch(void* const* d_in, const int* in_sizes, int n_in,
                              void* d_out, int out_size, void* d_ws,
                              size_t ws_size, hipStream_t stream) {
  (void)in_sizes;
  (void)n_in;
  (void)out_size;
  (void)ws_size;
  const float* query = (const float*)d_in[0];
  const float* attn_bias = (const float*)d_in[1];
  const float* Wq = (const float*)d_in[2];
  const float* bq = (const float*)d_in[3];
  const float* Wk = (const float*)d_in[4];
  const float* bk = (const float*)d_in[5];
  const float* Wv = (const float*)d_in[6];
  const float* bv = (const float*)d_in[7];
  const float* Wo = (const float*)d_in[8];
  const float* bo = (const float*)d_in[9];

  const size_t NTE = (size_t)T_DIM * B_DIM * E_DIM;  // 2M floats
  float* qs = (float*)d_ws;
  float* ks = qs + NTE;
  float* vs = ks + NTE;
  float* as = vs + NTE;  // 32 MB total workspace

  // 3 * (256*32) = 24576 tiles, 4 waves/block
  proj_qkv_kernel<<<6144, 128, 0, stream>>>(query, Wq, bq, Wk, bk, Wv, bv, qs,
                                            ks, vs);
  // 16 heads * 128 row-tiles = 2048 waves, 2 waves/block
  attn_kernel<<<1024, 64, 0, stream>>>(qs, ks, vs, attn_bias, as);
  // 256*32 = 8192 tiles
  proj_out_kernel<<<2048, 128, 0, stream>>>(as, Wo, bo, (float*)d_out);
}